// PVLoss_59210419142932
// MI455X (gfx1250) — compile-verified
//
#include <hip/hip_runtime.h>

// PV SAT-overlap loss on MI455X (gfx1250, wave32).
// One wave per box pair. D = corners(16x4) x axes(4x16) via one
// V_WMMA_F32_16X16X4_F32, so the 8-corner extent reduction is a pure
// in-lane register tree; only the tiny 9-axis reduction crosses lanes
// (DPP8 + immediate ds_swizzle).

typedef __attribute__((ext_vector_type(2))) float v2f;
typedef __attribute__((ext_vector_type(8))) float v8f;

// ---- DPP8 helpers: arbitrary permute within groups of 8 lanes (VALU only)
#define DPP8SEL(a,b,c,d,e,f,g,h) \
    ((a)|((b)<<3)|((c)<<6)|((d)<<9)|((e)<<12)|((f)<<15)|((g)<<18)|((h)<<21))
constexpr int kDppXor1 = DPP8SEL(1,0,3,2,5,4,7,6);
constexpr int kDppXor2 = DPP8SEL(2,3,0,1,6,7,4,5);
constexpr int kDppXor4 = DPP8SEL(4,5,6,7,0,1,2,3);

template <int SEL>
__device__ __forceinline__ float dpp8f(float v) {
    return __int_as_float(__builtin_amdgcn_mov_dpp8(__float_as_int(v), SEL));
}

// ---- ds_swizzle: group-of-32 mode, offset = {xor[14:10], or[9:5], and[4:0]}
constexpr int kSwzXor8  = (8  << 10) | 0x1F;   // lane ^= 8
constexpr int kSwzXor16 = (16 << 10) | 0x1F;   // lane ^= 16

template <int PAT>
__device__ __forceinline__ float swzf(float v) {
    return __int_as_float(__builtin_amdgcn_ds_swizzle(__float_as_int(v), PAT));
}

__global__ void pv_zero_out(float* out) { out[0] = 0.0f; }

__global__ __launch_bounds__(256)
void pv_sat_kernel(const float* __restrict__ bb,    // [B,6]
                   const float* __restrict__ rot,   // [B,3,3]
                   const float* __restrict__ trs,   // [B,3]
                   float* __restrict__ out,
                   int nB)
{
    const int lane = threadIdx.x & 31;
    const int wave = threadIdx.x >> 5;
    const int i = blockIdx.y;                 // first box
    const int j = blockIdx.x * 8 + wave;      // second box

    __shared__ float partial[8];

    float penalty = 0.0f;

    if (j > i && j < nB) {                    // wave-uniform: EXEC stays all-1s
        const int n  = lane & 15;             // row (corner) / col (axis) index
        const int hi = lane >> 4;             // 0: K=0,1 half  1: K=2,3 half

        // ------------- A operand: world-space corners (16 rows) ----------
        // rows 0-7 = corners of box i, rows 8-15 = corners of box j
        const int    bx = (n < 8) ? i : j;
        const int    k  = n & 7;              // corner enum: bit2->x, bit1->y, bit0->z
        const float* Rb = rot + bx * 9;
        const float* bv = bb  + bx * 6;
        const float* tb = trs + bx * 3;
        float cx = bv[0] + bv[3] * (float)((k >> 2) & 1);
        float cy = bv[1] + bv[4] * (float)((k >> 1) & 1);
        float cz = bv[2] + bv[5] * (float)(k & 1);
        // row-vector * R + t   (einsum 'bkd,bde->bke')
        float wx = cx * Rb[0] + cy * Rb[3] + cz * Rb[6] + tb[0];
        float wy = cx * Rb[1] + cy * Rb[4] + cz * Rb[7] + tb[1];
        float wz = cx * Rb[2] + cy * Rb[5] + cz * Rb[8] + tb[2];
        v2f Aop;
        Aop.x = hi ? wz : wx;                 // K=0 (x) | K=2 (z)
        Aop.y = hi ? 0.0f : wy;               // K=1 (y) | K=3 (pad)

        // ------------- B operand: 9 SAT axes (16 cols, zero padded) ------
        const float* Ri = rot + i * 9;
        const float* Rj = rot + j * 9;
        const int m = n;                      // axis column
        float ax = 0.0f, ay = 0.0f, az = 0.0f;
        if (m < 3) {                          // face axes = rows of R_i (I @ R)
            ax = Ri[m*3+0]; ay = Ri[m*3+1]; az = Ri[m*3+2];
        } else if (m < 9) {                   // cross combos (0,0)(0,1)(0,2)(1,1)(1,2)(2,2)
            const int t  = m - 3;
            const int ca = (t < 3) ? 0 : ((t < 5) ? 1 : 2);
            const int cb = (t < 3) ? t : ((t < 5) ? (t - 2) : 2);
            float ux = Ri[ca*3+0], uy = Ri[ca*3+1], uz = Ri[ca*3+2];
            float vx = Rj[cb*3+0], vy = Rj[cb*3+1], vz = Rj[cb*3+2];
            ax = uy*vz - uz*vy;
            ay = uz*vx - ux*vz;
            az = ux*vy - uy*vx;
        }
        v2f Bop;
        Bop.x = hi ? az : ax;                 // K=0 | K=2
        Bop.y = hi ? 0.0f : ay;               // K=1 | K=3 (pad)

        // exact-zero cross-axis invalidation (reference's torch.equal skip)
        bool partnz = (m >= 3) && (m < 9) &&
                      (hi ? (az != 0.0f) : (ax != 0.0f || ay != 0.0f));
        unsigned long long bal = __ballot(partnz);
        unsigned comb  = (unsigned)bal | (unsigned)(bal >> 16);
        unsigned vmask = 0x7u | (comb & 0x1F8u);   // bits 0..8: axis validity

        // ------------- D = corners x axes (16x16x4 f32 WMMA) -------------
        // lane L, reg r: D[r] = corner (r + 8*(L>=16)) . axis (L&15)
        v8f Cz = {0.f, 0.f, 0.f, 0.f, 0.f, 0.f, 0.f, 0.f};
        v8f D = __builtin_amdgcn_wmma_f32_16x16x4_f32(
                    false, Aop, false, Bop, (short)0, Cz, false, false);

        // ------------- extents: pure in-lane tree over 8 corners ---------
        // lanes 0-15: box i on axis n; lanes 16-31: box j on axis n
        float mn = D[0], mx = D[0];
        #pragma unroll
        for (int r = 1; r < 8; ++r) {
            mn = fminf(mn, D[r]);
            mx = fmaxf(mx, D[r]);
        }
        // swap box-i/box-j extents between wave halves
        float mno = swzf<kSwzXor16>(mn);
        float mxo = swzf<kSwzXor16>(mx);
        float total = fmaxf(mx, mxo) - fminf(mn, mno);
        float slen  = (mx - mn) + (mxo - mno);
        float ov    = slen - total;

        // ------------- SAT decision across 9 axes ------------------------
        unsigned use = (vmask >> n) & 1u;     // axes 9..15 auto-invalid
        unsigned long long failbal =
            __ballot(use && !(total < slen)); // any set bit -> separating axis
        float ovm = use ? ov : __builtin_inff();
        // min over the 16 axis lanes of each half (halves are identical)
        ovm = fminf(ovm, dpp8f<kDppXor1>(ovm));
        ovm = fminf(ovm, dpp8f<kDppXor2>(ovm));
        ovm = fminf(ovm, dpp8f<kDppXor4>(ovm));
        ovm = fminf(ovm, swzf<kSwzXor8>(ovm));
        penalty = (failbal == 0ull) ? ovm : 0.0f;
    }

    if (lane == 0) partial[wave] = penalty;
    __syncthreads();
    if (threadIdx.x == 0) {
        float s = 0.0f;
        #pragma unroll
        for (int w = 0; w < 8; ++w) s += partial[w];
        if (s != 0.0f) unsafeAtomicAdd(out, s);        // native global_atomic_add_f32
    }
}

extern "C" void kernel_launch(void* const* d_in, const int* in_sizes, int n_in,
                              void* d_out, int out_size, void* d_ws, size_t ws_size,
                              hipStream_t stream) {
    const float* bb  = (const float*)d_in[0];  // [B,6]
    const float* rot = (const float*)d_in[1];  // [B,3,3]
    const float* trs = (const float*)d_in[2];  // [B,1,3]
    float* out = (float*)d_out;

    const int nB = in_sizes[0] / 6;            // 1024

    pv_zero_out<<<1, 1, 0, stream>>>(out);

    dim3 grid((nB + 7) / 8, nB, 1);            // 8 waves (pairs j) per block, i per row
    pv_sat_kernel<<<grid, 256, 0, stream>>>(bb, rot, trs, out, nB);
}